// GATLayer_12567074308556
// MI455X (gfx1250) — compile-verified
//
#include <hip/hip_runtime.h>

typedef __attribute__((ext_vector_type(2))) float v2f;
typedef __attribute__((ext_vector_type(8))) float v8f;

#define TPB 256
#define NEG_SLOPE 0.2f

static inline unsigned gblocks(long n, int t) { return (unsigned)((n + t - 1) / t); }

__device__ __forceinline__ void atomicMaxF(float* addr, float val) {
  unsigned int* ua = (unsigned int*)addr;
  unsigned int cur = *ua;
  while (__uint_as_float(cur) < val) {
    unsigned int prev = atomicCAS(ua, cur, __float_as_uint(val));
    if (prev == cur) break;
    cur = prev;
  }
}

__global__ void fill_k(float* __restrict__ p, float v, long n) {
  long i = (long)blockIdx.x * blockDim.x + threadIdx.x;
  if (i < n) p[i] = v;
}

// ---- self-loop edge_attr = per-destination mean of incoming edge_attr ----
__global__ void loop_attr_accum_k(const int* __restrict__ dst, const float* __restrict__ ew,
                                  float* __restrict__ asum, float* __restrict__ cnt, long E) {
  long e = (long)blockIdx.x * blockDim.x + threadIdx.x;
  if (e >= E) return;
  int d = dst[e];
  atomicAdd(&asum[d], ew[e]);
  atomicAdd(&cnt[d], 1.0f);
}

__global__ void loop_attr_final_k(const float* __restrict__ asum, const float* __restrict__ cnt,
                                  float* __restrict__ la, long N) {
  long i = (long)blockIdx.x * blockDim.x + threadIdx.x;
  if (i < N) la[i] = asum[i] / fmaxf(cnt[i], 1.0f);
}

// a_edge per unit edge_attr: wedot[h] = sum_c W_edge[h*C+c]*att_edge[h*C+c]
__global__ void wedot_k(const float* __restrict__ W_edge, const float* __restrict__ att_edge,
                        float* __restrict__ wedot, int H, int C) {
  int h = threadIdx.x;
  if (h >= H) return;
  float s = 0.f;
  for (int c = 0; c < C; ++c) s += W_edge[h * C + c] * att_edge[h * C + c];
  wedot[h] = s;
}

// ---- D = A[M,K] x B[K,Ncol] via V_WMMA_F32_16X16X4_F32, one wave per 16x16 tile ----
// A frag (16x4): lanes 0-15 hold K = k+0,k+1; lanes 16-31 hold K = k+2,k+3 (ISA 7.12.2)
// B frag (4x16): VGPR v, lane-half g -> row K = k + v + 2g, col = lane%16
// C/D (16x16):   VGPR v, lane-half g -> row M = v + 8g, col = lane%16
__global__ void gemm_f32_wmma_k(const float* __restrict__ A, const float* __restrict__ B,
                                float* __restrict__ D, int M, int K, int Ncol) {
  int tm = blockIdx.x;
  int tn = blockIdx.y * blockDim.y + threadIdx.y;
  if (tn >= (Ncol >> 4)) return;  // wave-uniform guard: EXEC stays all-ones for WMMA
  int l = threadIdx.x;
  int g = l >> 4;
  int lm = l & 15;
  const float* Arow = A + (size_t)(tm * 16 + lm) * K;
  int col = tn * 16 + lm;
  v8f acc = {};
  for (int k = 0; k < K; k += 4) {
    v2f a, b;
    a.x = Arow[k + 2 * g + 0];
    a.y = Arow[k + 2 * g + 1];
    b.x = B[(size_t)(k + 2 * g + 0) * Ncol + col];
    b.y = B[(size_t)(k + 2 * g + 1) * Ncol + col];
    acc = __builtin_amdgcn_wmma_f32_16x16x4_f32(false, a, false, b, (short)0, acc,
                                                false, false);
  }
  for (int v = 0; v < 8; ++v)
    D[(size_t)(tm * 16 + v + 8 * g) * Ncol + col] = acc[v];
}

// per-(node,head) attention coefficients: asrc[n,h] = h[n,h,:].att_src[h,:]
__global__ void attn_coef_k(const float* __restrict__ h, const float* __restrict__ att_s,
                            const float* __restrict__ att_d, float* __restrict__ asrc,
                            float* __restrict__ adst, long N, int H, int C) {
  long t = (long)blockIdx.x * blockDim.x + threadIdx.x;
  if (t >= N * H) return;
  long n = t / H;
  int hh = (int)(t % H);
  const float* row = h + (size_t)n * H * C + (size_t)hh * C;
  float s = 0.f, d = 0.f;
  for (int c = 0; c < C; ++c) {
    float v = row[c];
    s += v * att_s[hh * C + c];
    d += v * att_d[hh * C + c];
  }
  asrc[t] = s;
  adst[t] = d;
}

// edge scores (incl. self loops at e>=E) + per-destination segment max
__global__ void alpha_k(const int* __restrict__ src, const int* __restrict__ dst,
                        const float* __restrict__ ew, const float* __restrict__ la,
                        const float* __restrict__ asrc, const float* __restrict__ adst,
                        const float* __restrict__ wedot, float* __restrict__ alpha,
                        float* __restrict__ amax, long E, long N, int H) {
  long e = (long)blockIdx.x * blockDim.x + threadIdx.x;
  if (e >= E + N) return;
  int s, d;
  float ea;
  if (e < E) { s = src[e]; d = dst[e]; ea = ew[e]; }
  else       { s = d = (int)(e - E); ea = la[e - E]; }
  for (int hh = 0; hh < H; ++hh) {
    float a = asrc[(size_t)s * H + hh] + adst[(size_t)d * H + hh] + ea * wedot[hh];
    a = a > 0.f ? a : NEG_SLOPE * a;
    alpha[e * H + hh] = a;
    atomicMaxF(&amax[(size_t)d * H + hh], a);
  }
}

// exp(alpha - amax[dst]) + per-destination segment sum (overwrites alpha with ex)
__global__ void ex_k(const int* __restrict__ dst, float* __restrict__ alpha,
                     const float* __restrict__ amax, float* __restrict__ denom,
                     long E, long N, int H) {
  long e = (long)blockIdx.x * blockDim.x + threadIdx.x;
  if (e >= E + N) return;
  int d = (e < E) ? dst[e] : (int)(e - E);
  for (int hh = 0; hh < H; ++hh) {
    float ex = __expf(alpha[e * H + hh] - amax[(size_t)d * H + hh]);
    alpha[e * H + hh] = ex;
    atomicAdd(&denom[(size_t)d * H + hh], ex);
  }
}

// weighted scatter-aggregate: acc[dst,h,c] += (ex/denom) * hin[src,h,c]; 4 channels/thread
__global__ void agg_k(const int* __restrict__ src, const int* __restrict__ dst,
                      const float* __restrict__ hin, const float* __restrict__ alpha,
                      const float* __restrict__ denom, float* __restrict__ acc,
                      long E, long N, int H, int C) {
  int HC = H * C;
  int CH4 = HC >> 2;
  long tid = (long)blockIdx.x * blockDim.x + threadIdx.x;
  long total = (E + N) * (long)CH4;
  if (tid >= total) return;
  long e = tid / CH4;
  int q = (int)(tid % CH4);
  int c0 = q * 4;
  int hh = c0 / C;
  int s, d;
  if (e < E) { s = src[e]; d = dst[e]; }
  else       { s = d = (int)(e - E); }
  float att = alpha[e * H + hh] / denom[(size_t)d * H + hh];
  float4 v = *(const float4*)(hin + (size_t)s * HC + c0);
  float* ap = acc + (size_t)d * HC + c0;
  atomicAdd(ap + 0, att * v.x);
  atomicAdd(ap + 1, att * v.y);
  atomicAdd(ap + 2, att * v.z);
  atomicAdd(ap + 3, att * v.w);
}

__global__ void bias_relu_inplace_k(float* __restrict__ a, const float* __restrict__ b,
                                    long N, int HC) {
  long i = (long)blockIdx.x * blockDim.x + threadIdx.x;
  if (i >= N * HC) return;
  float v = a[i] + b[i % HC];
  a[i] = v > 0.f ? v : 0.f;
}

__global__ void bias_inplace_k(float* __restrict__ a, const float* __restrict__ b,
                               long N, int HC) {
  long i = (long)blockIdx.x * blockDim.x + threadIdx.x;
  if (i >= N * HC) return;
  a[i] = a[i] + b[i % HC];
}

// ----------------- one full GATConv layer -----------------
static void run_gat_layer(const float* xin, int IN, const int* src, const int* dst,
                          const float* ew, const float* la, const float* Wmat,
                          const float* att_s, const float* att_d, const float* W_edge,
                          const float* att_e, long N, long E, int H, int C, float* h,
                          float* asrc, float* adst, float* wedot, float* alpha,
                          float* amax, float* denom, float* acc, hipStream_t stream) {
  int HC = H * C;
  // projection h = xin @ Wmat  (WMMA f32)
  dim3 gblk(32, 4);
  dim3 ggrid((unsigned)(N / 16), (unsigned)(((HC >> 4) + 3) / 4));
  gemm_f32_wmma_k<<<ggrid, gblk, 0, stream>>>(xin, Wmat, h, (int)N, IN, HC);
  wedot_k<<<1, 32, 0, stream>>>(W_edge, att_e, wedot, H, C);
  attn_coef_k<<<gblocks(N * H, TPB), TPB, 0, stream>>>(h, att_s, att_d, asrc, adst, N, H, C);
  fill_k<<<gblocks(N * H, TPB), TPB, 0, stream>>>(amax, -3.402823466e38f, N * H);
  fill_k<<<gblocks(N * H, TPB), TPB, 0, stream>>>(denom, 0.f, N * H);
  fill_k<<<gblocks(N * HC, TPB), TPB, 0, stream>>>(acc, 0.f, N * HC);
  alpha_k<<<gblocks(E + N, TPB), TPB, 0, stream>>>(src, dst, ew, la, asrc, adst, wedot,
                                                   alpha, amax, E, N, H);
  ex_k<<<gblocks(E + N, TPB), TPB, 0, stream>>>(dst, alpha, amax, denom, E, N, H);
  agg_k<<<gblocks((E + N) * (HC >> 2), TPB), TPB, 0, stream>>>(src, dst, h, alpha, denom,
                                                               acc, E, N, H, C);
}

extern "C" void kernel_launch(void* const* d_in, const int* in_sizes, int n_in,
                              void* d_out, int out_size, void* d_ws, size_t ws_size,
                              hipStream_t stream) {
  const float* x        = (const float*)d_in[0];
  const int*   ei       = (const int*)d_in[1];
  const float* ew       = (const float*)d_in[2];
  const float* W1       = (const float*)d_in[3];
  const float* att_src1 = (const float*)d_in[4];
  const float* att_dst1 = (const float*)d_in[5];
  const float* W_edge1  = (const float*)d_in[6];
  const float* att_e1   = (const float*)d_in[7];
  const float* b1       = (const float*)d_in[8];
  const float* W2       = (const float*)d_in[9];
  const float* att_src2 = (const float*)d_in[10];
  const float* att_dst2 = (const float*)d_in[11];
  const float* W_edge2  = (const float*)d_in[12];
  const float* att_e2   = (const float*)d_in[13];
  const float* b2       = (const float*)d_in[14];
  float* out = (float*)d_out;

  long N = in_sizes[0] / 128;  // 50000
  long E = in_sizes[1] / 2;    // 800000
  const int* src = ei;
  const int* dst = ei + E;

  // workspace carve-out (floats)
  float* W = (float*)d_ws;
  size_t off = 0;
  auto alloc = [&](size_t n) { float* p = W + off; off += n; return p; };
  float* h1       = alloc((size_t)N * 128);      // layer1 projection; reused for layer2
  float* acc1     = alloc((size_t)N * 128);      // layer1 aggregate -> relu'd features
  float* asrc     = alloc((size_t)N * 2);
  float* adst     = alloc((size_t)N * 2);
  float* amax     = alloc((size_t)N * 2);
  float* denom    = alloc((size_t)N * 2);
  float* alphabuf = alloc((size_t)(E + N) * 2);
  float* attr_sum = alloc((size_t)N);
  float* cnt      = alloc((size_t)N);
  float* la       = alloc((size_t)N);
  float* wedot    = alloc(4);
  (void)ws_size; (void)n_in; (void)out_size;

  // ---- self-loop attr (shared by both layers) ----
  fill_k<<<gblocks(N, TPB), TPB, 0, stream>>>(attr_sum, 0.f, N);
  fill_k<<<gblocks(N, TPB), TPB, 0, stream>>>(cnt, 0.f, N);
  loop_attr_accum_k<<<gblocks(E, TPB), TPB, 0, stream>>>(dst, ew, attr_sum, cnt, E);
  loop_attr_final_k<<<gblocks(N, TPB), TPB, 0, stream>>>(attr_sum, cnt, la, N);

  // ---- layer 1: heads=2, C=64, concat -> 128, then +b1, ReLU (in place in acc1) ----
  run_gat_layer(x, 128, src, dst, ew, la, W1, att_src1, att_dst1, W_edge1, att_e1,
                N, E, 2, 64, h1, asrc, adst, wedot, alphabuf, amax, denom, acc1, stream);
  bias_relu_inplace_k<<<gblocks(N * 128, TPB), TPB, 0, stream>>>(acc1, b1, N, 128);

  // ---- layer 2: heads=1, C=64, concat=False (mean over 1 head = identity), +b2 ----
  // aggregate directly into d_out (zeroed inside run_gat_layer); reuse h1 for projections
  run_gat_layer(acc1, 128, src, dst, ew, la, W2, att_src2, att_dst2, W_edge2, att_e2,
                N, E, 1, 64, h1, asrc, adst, wedot, alphabuf, amax, denom, out, stream);
  bias_inplace_k<<<gblocks(N * 64, TPB), TPB, 0, stream>>>(out, b2, N, 64);
}